// GatedMultiHeadSectionAttentionImputer_29764123361476
// MI455X (gfx1250) — compile-verified
//
#include <hip/hip_runtime.h>
#include <hip/hip_bf16.h>
#include <math.h>

typedef float v2f __attribute__((ext_vector_type(2)));
typedef float v8f __attribute__((ext_vector_type(8)));

#define WMMA4(a, b, c) \
  __builtin_amdgcn_wmma_f32_16x16x4_f32(false, (a), false, (b), (short)0, (c), false, false)

constexpr int S_ = 4096, E_ = 512, H_ = 8, DK_ = 64, TD_ = 512, M_ = 2048, N_ = 2048;

// ---------------------------------------------------------------------------
// Kernel 1: copy full ehr -> out (missing rows overwritten later)
// ---------------------------------------------------------------------------
__global__ void copy_kernel(const float* __restrict__ src, float* __restrict__ dst, int n4) {
  int i = blockIdx.x * blockDim.x + threadIdx.x;
  if (i < n4) {
    *(float4*)(dst + (size_t)i * 4) = *(const float4*)(src + (size_t)i * 4);
  }
}

// ---------------------------------------------------------------------------
// Kernel 2: QKV projection. blockIdx.z selects (Wq,missing)/(Wk,exist)/(Wv,exist)
// Block: 128 threads = 4 waves; block tile 64(M) x 64(N); K staged 32 at a time.
// W chunk stored TRANSPOSED in LDS (Wt[n][k], stride 36) so B-fragments are
// per-lane contiguous float2 -> single ds_load_b64, no register packing movs.
// Output head-major: buf[(h*M + m)*DK + d]
// ---------------------------------------------------------------------------
__global__ __launch_bounds__(128) void proj_kernel(
    const float* __restrict__ ehr, const int* __restrict__ midx, const int* __restrict__ eidx,
    const float* __restrict__ Wq, const float* __restrict__ bq,
    const float* __restrict__ Wk, const float* __restrict__ bk,
    const float* __restrict__ Wv, const float* __restrict__ bv,
    float* __restrict__ ws) {
  const int z = blockIdx.z;
  const float* Wmat = (z == 0) ? Wq : ((z == 1) ? Wk : Wv);
  const float* bias = (z == 0) ? bq : ((z == 1) ? bk : bv);
  const int* ridx = (z == 0) ? midx : eidx;
  float* dst = ws + (size_t)z * ((size_t)H_ * M_ * DK_);

  __shared__ float As[64 * 36];   // A chunk  [m][k], 64x32, stride 36
  __shared__ float Wt[64 * 36];   // W chunk transposed [n][k], 64x32, stride 36

  const int tid = threadIdx.x;
  const int lane = tid & 31;
  const int wv = tid >> 5;
  const int l16 = lane & 15;
  const int hi = lane >> 4;
  const int koff = hi * 2;

  const int m0 = blockIdx.x * 64;
  const int n0 = blockIdx.y * 64;

  v8f acc[4];
  for (int nt = 0; nt < 4; ++nt)
    for (int j = 0; j < 8; ++j) acc[nt][j] = 0.0f;

  for (int kc = 0; kc < E_; kc += 32) {
    __syncthreads();
    // stage A chunk: 64x32, float4 per slot (512 slots / 128 threads)
    for (int s = tid; s < 64 * 8; s += 128) {
      int row = s >> 3, c4 = (s & 7) * 4;
      int g = ridx[m0 + row];
      *(float4*)(As + row * 36 + c4) = *(const float4*)(ehr + (size_t)g * E_ + kc + c4);
      if (kc + 32 < E_ && c4 == 0)
        __builtin_prefetch(ehr + (size_t)g * E_ + kc + 32, 0, 1);
    }
    // stage W chunk transposed: read W[k][n..n+3], scatter to Wt[n+j][k]
    for (int s = tid; s < 32 * 16; s += 128) {
      int row = s >> 4, c4 = (s & 15) * 4;
      const float4 w4 = *(const float4*)(Wmat + (size_t)(kc + row) * TD_ + n0 + c4);
      Wt[(c4 + 0) * 36 + row] = w4.x;
      Wt[(c4 + 1) * 36 + row] = w4.y;
      Wt[(c4 + 2) * 36 + row] = w4.z;
      Wt[(c4 + 3) * 36 + row] = w4.w;
      if (kc + 32 < E_ && c4 == 0)
        __builtin_prefetch(Wmat + (size_t)(kc + 32 + row) * TD_ + n0, 0, 1);
    }
    __syncthreads();

    for (int kk = 0; kk < 32; kk += 4) {
      v2f a = *(const v2f*)(As + (wv * 16 + l16) * 36 + kk + koff);
      for (int nt = 0; nt < 4; ++nt) {
        v2f b = *(const v2f*)(Wt + (nt * 16 + l16) * 36 + kk + koff);
        acc[nt] = WMMA4(a, b, acc[nt]);
      }
    }
  }

  // epilogue: +bias, scatter to head-major layout
  for (int nt = 0; nt < 4; ++nt) {
    int n = n0 + nt * 16 + l16;
    float bb = bias[n];
    int h = n >> 6, d = n & 63;
    for (int j = 0; j < 8; ++j) {
      int m = m0 + wv * 16 + j + hi * 8;
      dst[((size_t)h * M_ + m) * DK_ + d] = acc[nt][j] + bb;
    }
  }
}

// ---------------------------------------------------------------------------
// Kernel 3: gate vectors. gq[h][m] = Q[h][m]·gate_W[h][:64] + gate_b[h]
//                         gk[h][n] = K[h][n]·gate_W[h][64:]
// ---------------------------------------------------------------------------
__global__ void gate_kernel(const float* __restrict__ Qb, const float* __restrict__ Kb,
                            const float* __restrict__ gw, const float* __restrict__ gb,
                            float* __restrict__ gq, float* __restrict__ gk) {
  int t = blockIdx.x * blockDim.x + threadIdx.x;
  int total = H_ * M_;
  if (t >= 2 * total) return;
  bool isK = t >= total;
  int i = isK ? (t - total) : t;
  int h = i / M_, m = i - h * M_;
  const float* src = (isK ? Kb : Qb) + ((size_t)h * M_ + m) * DK_;
  const float* w = gw + h * 2 * DK_ + (isK ? DK_ : 0);
  float s = 0.0f;
  for (int d = 0; d < DK_; ++d) s += src[d] * w[d];
  if (!isK) s += gb[h];
  (isK ? gk : gq)[h * M_ + m] = s;
}

// ---------------------------------------------------------------------------
// Kernel 4: gated attention, flash-style online softmax.
// Block = (64 query rows, 1 head); 4 waves, each owns 16 rows.
// Per 64-key chunk: S=Q K^T, T=Bm Be^T (WMMA), gate+scale, online softmax,
// P round-trip through LDS, O += P V (WMMA, V stored transposed in LDS).
// ---------------------------------------------------------------------------
__global__ __launch_bounds__(128) void attn_kernel(
    const float* __restrict__ Qb, const float* __restrict__ Kb, const float* __restrict__ Vb,
    const float* __restrict__ cooc, const int* __restrict__ midx, const int* __restrict__ eidx,
    const float* __restrict__ gq, const float* __restrict__ gk, float* __restrict__ out) {
  const int h = blockIdx.y;
  const int m0 = blockIdx.x * 64;
  const int tid = threadIdx.x;
  const int lane = tid & 31;
  const int wv = tid >> 5;
  const int l16 = lane & 15;
  const int hi = lane >> 4;
  const int koff = hi * 2;

  __shared__ float Ks[64 * 68];   // K chunk [n][d], stride 68
  __shared__ float Vt[64 * 68];   // V chunk TRANSPOSED [d][n], stride 68
  __shared__ float Bes[64 * 68];  // exist bias chunk [n][d]; reused as P[4][16*68]
  __shared__ float gks[64];

  // Q and missing-bias fragments held in registers for all chunks
  v2f qf[16], bmf[16];
  {
    const int mrow = m0 + wv * 16 + l16;
    const float* qp = Qb + ((size_t)h * M_ + mrow) * DK_;
    const int gmi = midx[mrow];
    const float* bp = cooc + ((size_t)h * S_ + gmi) * DK_;
    for (int i = 0; i < 16; ++i) {
      qf[i] = *(const v2f*)(qp + i * 4 + koff);
      bmf[i] = *(const v2f*)(bp + i * 4 + koff);
    }
  }
  float gqv[8];
  for (int j = 0; j < 8; ++j) gqv[j] = gq[h * M_ + m0 + wv * 16 + j + hi * 8];

  v8f o[4];
  float rmax[8], rsum[8];
  for (int dt = 0; dt < 4; ++dt)
    for (int j = 0; j < 8; ++j) o[dt][j] = 0.0f;
  for (int j = 0; j < 8; ++j) { rmax[j] = -3.4e38f; rsum[j] = 0.0f; }

  for (int n0 = 0; n0 < N_; n0 += 64) {
    __syncthreads();
    // stage K (row-major), V (transposed), exist-bias (row-major)
    for (int s = tid; s < 64 * 16; s += 128) {
      int row = s >> 4, c4 = (s & 15) * 4;
      *(float4*)(Ks + row * 68 + c4) =
          *(const float4*)(Kb + ((size_t)h * N_ + n0 + row) * DK_ + c4);
      const float4 v4 = *(const float4*)(Vb + ((size_t)h * N_ + n0 + row) * DK_ + c4);
      Vt[(c4 + 0) * 68 + row] = v4.x;
      Vt[(c4 + 1) * 68 + row] = v4.y;
      Vt[(c4 + 2) * 68 + row] = v4.z;
      Vt[(c4 + 3) * 68 + row] = v4.w;
      int ge = eidx[n0 + row];
      *(float4*)(Bes + row * 68 + c4) =
          *(const float4*)(cooc + ((size_t)h * S_ + ge) * DK_ + c4);
      // prefetch next chunk (global_prefetch_b8)
      if (n0 + 64 < N_ && c4 == 0) {
        __builtin_prefetch(Kb + ((size_t)h * N_ + n0 + 64 + row) * DK_, 0, 1);
        __builtin_prefetch(Vb + ((size_t)h * N_ + n0 + 64 + row) * DK_, 0, 1);
        __builtin_prefetch(cooc + ((size_t)h * S_ + 2048 + n0 + 64 + row) * DK_, 0, 1);
      }
    }
    if (tid < 64) gks[tid] = gk[h * N_ + n0 + tid];
    __syncthreads();

    // S = Q K^T, T = Bm Be^T  (16 k-steps x 4 n-tiles each)
    v8f sa[4], ta[4];
    for (int nt = 0; nt < 4; ++nt)
      for (int j = 0; j < 8; ++j) { sa[nt][j] = 0.0f; ta[nt][j] = 0.0f; }
    for (int kk = 0; kk < 16; ++kk) {
      v2f a = qf[kk];
      v2f ab = bmf[kk];
      for (int nt = 0; nt < 4; ++nt) {
        v2f b = *(const v2f*)(Ks + (nt * 16 + l16) * 68 + kk * 4 + koff);
        sa[nt] = WMMA4(a, b, sa[nt]);
        v2f bb = *(const v2f*)(Bes + (nt * 16 + l16) * 68 + kk * 4 + koff);
        ta[nt] = WMMA4(ab, bb, ta[nt]);
      }
    }

    // gate + scale, chunk row-max (rows live in vector comp j + lane half)
    float gkv[4];
    for (int nt = 0; nt < 4; ++nt) gkv[nt] = gks[nt * 16 + l16];
    float cmax[8];
    for (int j = 0; j < 8; ++j) cmax[j] = -3.4e38f;
    for (int nt = 0; nt < 4; ++nt)
      for (int j = 0; j < 8; ++j) {
        float g = 0.5f + 0.5f * tanhf(gqv[j] + gkv[nt]);
        float v = sa[nt][j] * 0.125f + g * ta[nt][j];
        sa[nt][j] = v;
        cmax[j] = fmaxf(cmax[j], v);
      }
    float alpha[8];
    for (int j = 0; j < 8; ++j) {
      float v = cmax[j];
      for (int ofs = 1; ofs < 16; ofs <<= 1) v = fmaxf(v, __shfl_xor(v, ofs, 16));
      float mn = fmaxf(rmax[j], v);
      alpha[j] = __expf(rmax[j] - mn);
      rmax[j] = mn;
    }
    float psum[8];
    for (int j = 0; j < 8; ++j) psum[j] = 0.0f;
    for (int nt = 0; nt < 4; ++nt)
      for (int j = 0; j < 8; ++j) {
        float p = __expf(sa[nt][j] - rmax[j]);
        sa[nt][j] = p;
        psum[j] += p;
      }
    for (int j = 0; j < 8; ++j) {
      float v = psum[j];
      for (int ofs = 1; ofs < 16; ofs <<= 1) v += __shfl_xor(v, ofs, 16);
      rsum[j] = rsum[j] * alpha[j] + v;
      for (int dt = 0; dt < 4; ++dt) o[dt][j] *= alpha[j];
    }

    // write P into LDS (overlay on Bes; all Bes fragment reads are done)
    __syncthreads();
    float* Pw = Bes + wv * (16 * 68);
    for (int nt = 0; nt < 4; ++nt)
      for (int j = 0; j < 8; ++j)
        Pw[(j + hi * 8) * 68 + nt * 16 + l16] = sa[nt][j];
    asm volatile("s_wait_dscnt 0" ::: "memory");  // same-wave LDS store->fragment load

    // O += P V   (B-frags contiguous from transposed Vt)
    for (int kk = 0; kk < 16; ++kk) {
      v2f a = *(const v2f*)(Pw + l16 * 68 + kk * 4 + koff);
      for (int dt = 0; dt < 4; ++dt) {
        v2f b = *(const v2f*)(Vt + (dt * 16 + l16) * 68 + kk * 4 + koff);
        o[dt] = WMMA4(a, b, o[dt]);
      }
    }
  }

  // finalize: O /= rowsum, scatter to out[missing_row][h*64+d]
  for (int j = 0; j < 8; ++j) {
    int m = m0 + wv * 16 + j + hi * 8;
    int grow = midx[m];
    float inv = 1.0f / rsum[j];
    for (int dt = 0; dt < 4; ++dt) {
      int d = dt * 16 + l16;
      out[(size_t)grow * TD_ + h * DK_ + d] = o[dt][j] * inv;
    }
  }
}

// ---------------------------------------------------------------------------
extern "C" void kernel_launch(void* const* d_in, const int* in_sizes, int n_in,
                              void* d_out, int out_size, void* d_ws, size_t ws_size,
                              hipStream_t stream) {
  (void)in_sizes; (void)n_in; (void)out_size; (void)ws_size;
  const float* ehr  = (const float*)d_in[0];
  const int*   eidx = (const int*)d_in[1];
  const int*   midx = (const int*)d_in[2];
  const float* Wq   = (const float*)d_in[3];
  const float* bq   = (const float*)d_in[4];
  const float* Wk   = (const float*)d_in[5];
  const float* bk   = (const float*)d_in[6];
  const float* Wv   = (const float*)d_in[7];
  const float* bv   = (const float*)d_in[8];
  const float* cooc = (const float*)d_in[9];
  const float* gw   = (const float*)d_in[10];
  const float* gb   = (const float*)d_in[11];
  float* out = (float*)d_out;
  float* ws  = (float*)d_ws;

  float* Qb = ws;
  float* Kb = ws + (size_t)H_ * M_ * DK_;
  float* Vb = ws + 2 * (size_t)H_ * M_ * DK_;
  float* gq = ws + 3 * (size_t)H_ * M_ * DK_;
  float* gk = gq + H_ * M_;

  copy_kernel<<<(S_ * E_ / 4 + 255) / 256, 256, 0, stream>>>(ehr, out, S_ * E_ / 4);
  proj_kernel<<<dim3(M_ / 64, TD_ / 64, 3), 128, 0, stream>>>(
      ehr, midx, eidx, Wq, bq, Wk, bk, Wv, bv, ws);
  gate_kernel<<<(2 * H_ * M_ + 255) / 256, 256, 0, stream>>>(Qb, Kb, gw, gb, gq, gk);
  attn_kernel<<<dim3(M_ / 64, H_), 128, 0, stream>>>(
      Qb, Kb, Vb, cooc, midx, eidx, gq, gk, out);
}